// QuanvolutionEnhanced_65481071407599
// MI455X (gfx1250) — compile-verified
//
#include <hip/hip_runtime.h>
#include <math.h>

typedef __attribute__((ext_vector_type(2))) float v2f;
typedef __attribute__((ext_vector_type(8))) float v8f;

#define QN_LAYERS 2

// Apply RY(angle) on the qubit whose index bit-weight is STRIDE (8,4,2,1),
// across a fully register-resident 16-element real statevector.
template <int STRIDE>
__device__ __forceinline__ void ry_all(float st[16], float c, float s) {
#pragma unroll
  for (int i = 0; i < 16; ++i) {
    if ((i & STRIDE) == 0) {
      float s0 = st[i], s1 = st[i + STRIDE];
      st[i] = c * s0 - s * s1;
      st[i + STRIDE] = s * s0 + c * s1;
    }
  }
}

__global__ __launch_bounds__(256) void quanv_fused_kernel(
    const float* __restrict__ x,     // [nb, 784]
    const float* __restrict__ wts,   // [8]
    const float* __restrict__ lin_w, // [10, 784] row-major
    const float* __restrict__ lin_b, // [10]
    float* __restrict__ out,         // [nb, 10] log-softmax
    int nb) {
  __shared__ float feats[16][784];   // 49 KB: per-block feature tile
  __shared__ float part[8][16][16];  // 8 KB: split-K partial C tiles
  __shared__ float logits[16][16];   // 1 KB

  const int b0 = (int)blockIdx.x * 16;
  const int t = (int)threadIdx.x;

  // Uniform shared-rotation cos/sin (scalar loads; 8 values).
  float cw[8], sw[8];
#pragma unroll
  for (int i = 0; i < 8; ++i) {
    float h = 0.5f * wts[i];
    sincosf(h, &sw[i], &cw[i]);
  }

  // ---------------- Phase A: quantum feature map into LDS ----------------
  for (int p = t; p < 16 * 196; p += 256) {
    int r = p / 196;            // row within block tile
    int q = p - r * 196;        // patch index 0..195
    int b = b0 + r;
    int pi = q / 14, pj = q - pi * 14;

    float a0 = 0.f, a1 = 0.f, a2 = 0.f, a3 = 0.f;
    if (b < nb) {
      const float* row = x + (size_t)b * 784 + (size_t)(2 * pi) * 28 + (size_t)(2 * pj);
      a0 = row[0]; a1 = row[1]; a2 = row[28]; a3 = row[29];
    }
    float c0, s0, c1, s1, c2, s2, c3, s3;
    sincosf(0.5f * a0, &s0, &c0);
    sincosf(0.5f * a1, &s1, &c1);
    sincosf(0.5f * a2, &s2, &c2);
    sincosf(0.5f * a3, &s3, &c3);

    // Product state after the 4 input RYs (closed form).
    float st[16];
#pragma unroll
    for (int k = 0; k < 16; ++k) {
      st[k] = ((k & 8) ? s0 : c0) * ((k & 4) ? s1 : c1) *
              ((k & 2) ? s2 : c2) * ((k & 1) ? s3 : c3);
    }

#pragma unroll
    for (int l = 0; l < QN_LAYERS; ++l) {
      ry_all<8>(st, cw[l * 4 + 0], sw[l * 4 + 0]);
      ry_all<4>(st, cw[l * 4 + 1], sw[l * 4 + 1]);
      ry_all<2>(st, cw[l * 4 + 2], sw[l * 4 + 2]);
      ry_all<1>(st, cw[l * 4 + 3], sw[l * 4 + 3]);
      // CNOT(q0,q1): where bit8==1, swap bit4
#pragma unroll
      for (int k = 0; k < 4; ++k) {
        float tmp = st[8 + k]; st[8 + k] = st[12 + k]; st[12 + k] = tmp;
      }
      // CNOT(q2,q3): where bit2==1, swap bit1
#pragma unroll
      for (int k = 0; k < 4; ++k) {
        float tmp = st[4 * k + 2]; st[4 * k + 2] = st[4 * k + 3]; st[4 * k + 3] = tmp;
      }
    }

    float z0 = 0.f, z1 = 0.f, z2 = 0.f, z3 = 0.f;
#pragma unroll
    for (int k = 0; k < 16; ++k) {
      float pk = st[k] * st[k];
      z0 += (k & 8) ? -pk : pk;
      z1 += (k & 4) ? -pk : pk;
      z2 += (k & 2) ? -pk : pk;
      z3 += (k & 1) ? -pk : pk;
    }
    feats[r][q * 4 + 0] = z0;
    feats[r][q * 4 + 1] = z1;
    feats[r][q * 4 + 2] = z2;
    feats[r][q * 4 + 3] = z3;
  }
  __syncthreads();

  // ---------------- Phase B: fp32 WMMA GEMM head, split-K over 8 waves ----
  // Each wave handles K-steps kk = wave, wave+8, ... (196 steps of K=4 total).
  // Fixed 25-iteration trip count for ALL waves (tail masked to zero via the
  // A fragment) so the loop fully unrolls into a straight-line WMMA stream.
  const int lane = t & 31;
  const int wave = t >> 5;
  const int mn = lane & 15;  // A-row index == B/C column index for this lane
  const int kh = lane >> 4;  // K half-pair select (A/B: K = k0 + 2*kh + {0,1})

  // Hoisted branch-free B column handling (zero-pad classes 10..15).
  const int bn = (mn < 10) ? mn : 0;
  const float bmask = (mn < 10) ? 1.f : 0.f;
  const float* __restrict__ brow = lin_w + (size_t)bn * 784;

  v8f acc = {};
#pragma unroll
  for (int it = 0; it < 25; ++it) {
    const int kk = wave + 8 * it;
    const bool live = (kk < 196);
    const float amask = live ? 1.f : 0.f;
    const int k0 = (live ? kk * 4 : 0) + kh * 2;  // always in-bounds

    // A fragment: rows m=mn of feats, K = k0,k0+1 (8B-aligned ds_load_b64)
    float2 af = *(const float2*)&feats[mn][k0];
    v2f a; a[0] = af.x * amask; a[1] = af.y * amask;
    // B fragment: column n=mn of lin_w^T (8B-aligned global_load_b64, L2-hot)
    float2 bf = *(const float2*)(brow + k0);
    v2f bv; bv[0] = bf.x * bmask; bv[1] = bf.y * bmask;

    acc = __builtin_amdgcn_wmma_f32_16x16x4_f32(
        /*neg_a=*/false, a, /*neg_b=*/false, bv,
        /*c_mod=*/(short)0, acc, /*reuse_a=*/false, /*reuse_b=*/false);
  }
  // C/D layout: VGPR v, lanes 0-15 -> M=v, lanes 16-31 -> M=v+8; N=lane%16
#pragma unroll
  for (int v = 0; v < 8; ++v) part[wave][v + 8 * kh][mn] = acc[v];
  __syncthreads();

  // Reduce split-K partials + bias.
  {
    int r = t >> 4, n = t & 15;
    float sum = 0.f;
#pragma unroll
    for (int w = 0; w < 8; ++w) sum += part[w][r][n];
    if (n < 10) sum += lin_b[n];
    logits[r][n] = sum;
  }
  __syncthreads();

  // Per-row log-softmax over the 10 valid classes.
  if (t < 16) {
    int b = b0 + t;
    if (b < nb) {
      float mx = logits[t][0];
#pragma unroll
      for (int n = 1; n < 10; ++n) mx = fmaxf(mx, logits[t][n]);
      float se = 0.f;
#pragma unroll
      for (int n = 0; n < 10; ++n) se += expf(logits[t][n] - mx);
      float lz = mx + logf(se);
#pragma unroll
      for (int n = 0; n < 10; ++n) out[(size_t)b * 10 + n] = logits[t][n] - lz;
    }
  }
}

extern "C" void kernel_launch(void* const* d_in, const int* in_sizes, int n_in,
                              void* d_out, int out_size, void* d_ws, size_t ws_size,
                              hipStream_t stream) {
  const float* x = (const float*)d_in[0];      // [bsz, 784]
  const float* wts = (const float*)d_in[1];    // [8]
  const float* lin_w = (const float*)d_in[2];  // [10, 784]
  const float* lin_b = (const float*)d_in[3];  // [10]
  float* out = (float*)d_out;                  // [bsz, 10]

  int nb = in_sizes[0] / 784;
  int blocks = (nb + 15) / 16;
  quanv_fused_kernel<<<blocks, 256, 0, stream>>>(x, wts, lin_w, lin_b, out, nb);
}